// EMA_19490561590048
// MI455X (gfx1250) — compile-verified
//
#include <hip/hip_runtime.h>

typedef __attribute__((ext_vector_type(2))) float v2f;
typedef __attribute__((ext_vector_type(8))) float v8f;

// ---- problem constants (from reference) ----
constexpr float kAlpha    = 0.9f;
constexpr float kOmAlpha  = 1.0f - 0.9f;   // 0.1
constexpr float kInvAlpha = 1.0f / 0.9f;

constexpr int B_ = 8, T_ = 4096, F_ = 1024;
constexpr int SEG     = 8;            // parallel segments along T
constexpr int SEG_LEN = T_ / SEG;     // 512
constexpr int WARM    = 256;          // alpha^256 ~ 2e-12 << fp32 eps -> exact at fp32
constexpr int FB      = F_ / 16;      // 64 feature blocks of 16
constexpr int PF_DIST = 4;            // prefetch distance in tiles (4*16 rows ahead)

constexpr float cpow(float a, int n) {
  float r = 1.0f;
  for (int i = 0; i < n; ++i) r *= a;
  return r;
}

// One wave (32 threads) per (batch, 16-feature block, segment).
// Tile = 16 timesteps x 16 features: Y = L*X + alpha^{t+1} (x) carry
// via 4 chained V_WMMA_F32_16X16X4_F32. Carry propagated with ds_bpermute.
// Software-pipelined: tile i+1's loads are issued before tile i's WMMA chain.
__global__ __launch_bounds__(32) void ema_wmma_kernel(const float* __restrict__ x,
                                                      float* __restrict__ y) {
  const int wid = blockIdx.x;
  const int fb  = wid % FB;
  const int seg = (wid / FB) % SEG;
  const int b   = wid / (FB * SEG);

  const int lane = threadIdx.x & 31;
  const int half = lane >> 4;   // 0: lanes 0-15, 1: lanes 16-31
  const int fl   = lane & 15;   // feature-in-block / matrix row M

  const int baseBF = b * T_ * F_ + fb * 16 + fl;

  // ---- exact alpha^M per lane (binary expansion, no transcendentals) ----
  float aM = 1.0f, p = kAlpha;
#pragma unroll
  for (int bit = 0; bit < 4; ++bit) {
    aM *= (fl & (1 << bit)) ? p : 1.0f;
    p *= p;
  }

  // ---- A-matrix (constant lower-triangular decay L), 16x4 chunks k=0..3 ----
  // A layout (32-bit 16x4): VGPR j, lane-half h -> column c = 4k + 2h + j, row M = lane%16
  constexpr float CINV[16] = {
      cpow(kInvAlpha, 0),  cpow(kInvAlpha, 1),  cpow(kInvAlpha, 2),  cpow(kInvAlpha, 3),
      cpow(kInvAlpha, 4),  cpow(kInvAlpha, 5),  cpow(kInvAlpha, 6),  cpow(kInvAlpha, 7),
      cpow(kInvAlpha, 8),  cpow(kInvAlpha, 9),  cpow(kInvAlpha, 10), cpow(kInvAlpha, 11),
      cpow(kInvAlpha, 12), cpow(kInvAlpha, 13), cpow(kInvAlpha, 14), cpow(kInvAlpha, 15)};
  const float inv2h = half ? (kInvAlpha * kInvAlpha) : 1.0f;

  v2f Amat[4];
#pragma unroll
  for (int k = 0; k < 4; ++k) {
#pragma unroll
    for (int j = 0; j < 2; ++j) {
      const int c = 4 * k + 2 * half + j;
      Amat[k][j] = (fl >= c) ? (kOmAlpha * aM * CINV[4 * k + j] * inv2h) : 0.0f;
    }
  }

  // ---- carry injection constants: C[row][f] = carry[f] * alpha^{row+1} ----
  // C/D layout: VGPR r holds rows {r (lanes 0-15), r+8 (lanes 16-31)}
  constexpr float APOW[8] = {cpow(kAlpha, 0), cpow(kAlpha, 1), cpow(kAlpha, 2),
                             cpow(kAlpha, 3), cpow(kAlpha, 4), cpow(kAlpha, 5),
                             cpow(kAlpha, 6), cpow(kAlpha, 7)};
  const float sel = half ? cpow(kAlpha, 9) : kAlpha;  // alpha^{1} or alpha^{9}

  // carry[f] lives in acc[7] of lanes 16-31 (row M=15); broadcast lane (fl+16) -> all
  const int permIdx = (fl + 16) << 2;

  int t0              = seg * SEG_LEN - (seg ? WARM : 0);
  const int nTiles    = (seg ? (SEG_LEN + WARM) : SEG_LEN) / 16;
  const int storeFrom = seg ? (WARM / 16) : 0;

  // B operand row offset within a tile for this lane: K = 4k + 2h + j
  const int rowOff = 2 * half;

  float carry = 0.0f;

  // ---- prologue: load tile 0's B operand ----
  float bcur[4][2];
#pragma unroll
  for (int k = 0; k < 4; ++k) {
#pragma unroll
    for (int j = 0; j < 2; ++j) {
      bcur[k][j] = x[baseBF + (t0 + 4 * k + rowOff + j) * F_];
    }
  }

  for (int tile = 0; tile < nTiles; ++tile, t0 += 16) {
    // ---- issue next tile's loads first (independent of carry / WMMA chain) ----
    float bnext[4][2];
    if (tile + 1 < nTiles) {
#pragma unroll
      for (int k = 0; k < 4; ++k) {
#pragma unroll
        for (int j = 0; j < 2; ++j) {
          bnext[k][j] = x[baseBF + (t0 + 16 + 4 * k + rowOff + j) * F_];
        }
      }
    }

    // deep prefetch of the stream (WGP scope; uniform branch keeps EXEC full)
    if (tile + PF_DIST < nTiles) {
      __builtin_prefetch(&x[baseBF + (t0 + PF_DIST * 16 + fl) * F_], 0, 3);
    }

    // C init with carry term
    v8f acc;
#pragma unroll
    for (int r = 0; r < 8; ++r) acc[r] = carry * (sel * APOW[r]);

    // Y = L * X + C  via 4 chained f32 WMMAs over K
#pragma unroll
    for (int k = 0; k < 4; ++k) {
      v2f Bm;
      Bm[0] = bcur[k][0];
      Bm[1] = bcur[k][1];
      acc = __builtin_amdgcn_wmma_f32_16x16x4_f32(false, Amat[k], false, Bm,
                                                  (short)0, acc, false, false);
    }

    // carry for next tile = Y[15][*]  (acc[7], lanes 16-31) -> broadcast to all
    carry = __builtin_bit_cast(
        float, __builtin_amdgcn_ds_bpermute(permIdx, __builtin_bit_cast(int, acc[7])));

    // store 16x16 tile (skip warm-up tiles); coalesced 64B rows
    if (tile >= storeFrom) {
#pragma unroll
      for (int r = 0; r < 8; ++r) {
        y[baseBF + (t0 + r + 8 * half) * F_] = acc[r];
      }
    }

    // rotate pipeline registers
#pragma unroll
    for (int k = 0; k < 4; ++k) {
#pragma unroll
      for (int j = 0; j < 2; ++j) bcur[k][j] = bnext[k][j];
    }
  }
}

extern "C" void kernel_launch(void* const* d_in, const int* in_sizes, int n_in,
                              void* d_out, int out_size, void* d_ws, size_t ws_size,
                              hipStream_t stream) {
  (void)in_sizes; (void)n_in; (void)d_ws; (void)ws_size; (void)out_size;
  const float* x = (const float*)d_in[0];
  float* y = (float*)d_out;

  const int grid = B_ * SEG * FB;  // 8 * 8 * 64 = 4096 waves
  ema_wmma_kernel<<<dim3(grid), dim3(32), 0, stream>>>(x, y);
}